// Readout_basis_16449724744560
// MI455X (gfx1250) — compile-verified
//
#include <hip/hip_runtime.h>
#include <hip/hip_bf16.h>

// CDNA5 wave32 WMMA types
typedef __attribute__((ext_vector_type(2))) float v2f;
typedef __attribute__((ext_vector_type(8))) float v8f;

#define MROWS 2048   // B*N = 8*256
#define FD    128    // F == S == 128

// ---------------------------------------------------------------------------
// Precompute vl[f] = sum_s W5[s]*W4[s, f], vr[f] = sum_s W5[s]*W4[s, 128+f],
// C = sum_s b4[s]*W5[s] + b5.  (Folds Wl/Wr @ W5^T: final matmul is linear.)
// ---------------------------------------------------------------------------
__global__ __launch_bounds__(128) void precompute_vecs(
    const float* __restrict__ W4, const float* __restrict__ b4,
    const float* __restrict__ W5, const float* __restrict__ b5,
    float* __restrict__ vl, float* __restrict__ vr, float* __restrict__ Cs) {
  const int f = threadIdx.x;  // 0..127
  float sl = 0.f, sr = 0.f;
  #pragma unroll 8
  for (int s = 0; s < 128; ++s) {
    const float w5 = W5[s];
    sl += w5 * W4[s * 256 + f];
    sr += w5 * W4[s * 256 + 128 + f];
  }
  vl[f] = sl;
  vr[f] = sr;
  if (f == 0) {
    float c = b5[0];
    for (int s = 0; s < 128; ++s) c += b4[s] * W5[s];
    Cs[0] = c;
  }
}

// ---------------------------------------------------------------------------
// Hout(2048x128) = Hin(2048x128) @ W^T(128x128) + bias, fp32 WMMA 16x16x4.
// One wave32 owns one 16x16 output tile; block = 8 waves covers full N=128
// for a 16-row stripe; grid.x = 128 stripes.
//
// A (16x4 f32): lane<16  -> M=lane,    VGPR0=K0, VGPR1=K1
//               lane>=16 -> M=lane-16, VGPR0=K2, VGPR1=K3
// B (4x16 f32): lane<16  -> N=lane,    VGPR0=K0, VGPR1=K1
//               lane>=16 -> N=lane-16, VGPR0=K2, VGPR1=K3
// D (16x16):    acc[g]: M = g + (lane>=16 ? 8 : 0), N = lane&15
// ---------------------------------------------------------------------------
__global__ __launch_bounds__(256) void gemm_wmma_f32(
    const float* __restrict__ Hin, const float* __restrict__ W,
    const float* __restrict__ bias, float* __restrict__ Hout) {
  const int lane = threadIdx.x & 31;
  const int wave = threadIdx.x >> 5;          // 0..7 -> n-tile
  const int m0   = blockIdx.x * 16;           // row stripe
  const int n0   = wave * 16;
  const int half = lane >> 4;                 // 0 or 1
  const int l15  = lane & 15;

  // Per-lane base pointers; +k0 walks K.  half*2 selects K pair {0,1}/{2,3}.
  const float* pA = Hin + (m0 + l15) * FD + half * 2;   // A[M][K]
  const float* pB = W   + (n0 + l15) * FD + half * 2;   // B[K][N] = W[N][K]

  v8f acc = {};
  #pragma unroll
  for (int k0 = 0; k0 < FD; k0 += 4) {
    const float2 av = *(const float2*)(pA + k0);
    const float2 bv = *(const float2*)(pB + k0);
    v2f a; a.x = av.x; a.y = av.y;
    v2f b; b.x = bv.x; b.y = bv.y;
    // (neg_a, A, neg_b, B, c_mod, C, reuse_a, reuse_b)
    acc = __builtin_amdgcn_wmma_f32_16x16x4_f32(
        false, a, false, b, (short)0, acc, false, false);
  }

  const int   col = n0 + l15;
  const float bv  = bias[col];
  #pragma unroll
  for (int g = 0; g < 8; ++g) {
    const int row = m0 + g + half * 8;
    Hout[row * FD + col] = acc[g] + bv;
  }
}

// ---------------------------------------------------------------------------
// ul[m] = h3[m] . vl ; ur[m] = h3[m] . vr.  One wave per row, 4 cols/lane.
// ---------------------------------------------------------------------------
__global__ __launch_bounds__(256) void row_dots(
    const float* __restrict__ H, const float* __restrict__ vl,
    const float* __restrict__ vr, float* __restrict__ ul,
    float* __restrict__ ur) {
  const int lane = threadIdx.x & 31;
  const int wave = threadIdx.x >> 5;
  const int row  = blockIdx.x * 8 + wave;

  const float4 h4 = *(const float4*)(H + row * FD + lane * 4);
  const float4 l4 = *(const float4*)(vl + lane * 4);
  const float4 r4 = *(const float4*)(vr + lane * 4);
  float sl = h4.x * l4.x + h4.y * l4.y + h4.z * l4.z + h4.w * l4.w;
  float sr = h4.x * r4.x + h4.y * r4.y + h4.z * r4.z + h4.w * r4.w;
  #pragma unroll
  for (int off = 16; off > 0; off >>= 1) {
    sl += __shfl_down(sl, off, 32);
    sr += __shfl_down(sr, off, 32);
  }
  if (lane == 0) { ul[row] = sl; ur[row] = sr; }
}

// ---------------------------------------------------------------------------
// out[b*65536 + i*256 + j] = ul[b*256+i] + ur[b*256+j] + C
// ---------------------------------------------------------------------------
__global__ __launch_bounds__(256) void expand_out(
    const float* __restrict__ ul, const float* __restrict__ ur,
    const float* __restrict__ Cs, float* __restrict__ out) {
  const unsigned idx = blockIdx.x * 256u + threadIdx.x;  // 0..524287
  const unsigned j = idx & 255u;
  const unsigned i = (idx >> 8) & 255u;
  const unsigned b = idx >> 16;
  out[idx] = ul[b * 256u + i] + ur[b * 256u + j] + Cs[0];
}

extern "C" void kernel_launch(void* const* d_in, const int* in_sizes, int n_in,
                              void* d_out, int out_size, void* d_ws, size_t ws_size,
                              hipStream_t stream) {
  (void)in_sizes; (void)n_in; (void)out_size; (void)ws_size;
  const float* x  = (const float*)d_in[0];
  const float* W1 = (const float*)d_in[1];
  const float* b1 = (const float*)d_in[2];
  const float* W2 = (const float*)d_in[3];
  const float* b2 = (const float*)d_in[4];
  const float* W3 = (const float*)d_in[5];
  const float* b3 = (const float*)d_in[6];
  const float* W4 = (const float*)d_in[7];
  const float* b4 = (const float*)d_in[8];
  const float* W5 = (const float*)d_in[9];
  const float* b5 = (const float*)d_in[10];
  float* out = (float*)d_out;

  // Workspace layout (floats): ~2.1 MB total
  float* ws   = (float*)d_ws;
  float* bufA = ws;                         // 2048*128
  float* bufB = bufA + MROWS * FD;          // 2048*128
  float* vl   = bufB + MROWS * FD;          // 128   (16B aligned)
  float* vr   = vl + FD;                    // 128   (16B aligned)
  float* Cs   = vr + FD;                    // 1
  float* ul   = Cs + 1;                     // 2048
  float* ur   = ul + MROWS;                 // 2048

  precompute_vecs<<<1, 128, 0, stream>>>(W4, b4, W5, b5, vl, vr, Cs);

  gemm_wmma_f32<<<MROWS / 16, 256, 0, stream>>>(x,    W1, b1, bufA);  // h1
  gemm_wmma_f32<<<MROWS / 16, 256, 0, stream>>>(bufA, W2, b2, bufB);  // h2
  gemm_wmma_f32<<<MROWS / 16, 256, 0, stream>>>(bufB, W3, b3, bufA);  // h3

  row_dots<<<MROWS / 8, 256, 0, stream>>>(bufA, vl, vr, ul, ur);

  expand_out<<<(8 * 256 * 256) / 256, 256, 0, stream>>>(ul, ur, Cs, out);
}